// GATNet_90993177133850
// MI455X (gfx1250) — compile-verified
//
#include <hip/hip_runtime.h>
#include <hip/hip_bf16.h>

// N=50000 nodes, E=800000 edges, D=64, H=2, C=64, H*C=128.
// Memory-bound GAT layer: only x_src = right @ W_lin is a real GEMM;
// attention logits collapse to F @ (W . att) matvecs (saves ~1.2GB traffic).
// All GEMMs use V_WMMA_F32_16X16X4_F32 (full fp32 precision matrix core).

typedef __attribute__((ext_vector_type(2))) float v2f;
typedef __attribute__((ext_vector_type(8))) float v8f;

#define LB256 __launch_bounds__(256)

// ---------------------------------------------------------------------------
// Generic fp32 WMMA GEMM: Y[M,NCOLS] = act(X)[M,K] @ W[K,NCOLS] (+ bias)
// SPLIT: X = concat(X0[M,64], X1[M,64]) along K (K must be 128).
// Block = 256 threads = 8 waves. NCOLS=128 -> 8 col tiles, 16 rows/block.
// NCOLS=64 -> 4 col tiles x 2 row tiles, 32 rows/block.
// ---------------------------------------------------------------------------
template <int K, int NCOLS, bool RELUX, bool SPLIT>
__global__ LB256 void gemm_wmma_kernel(const float* __restrict__ X0,
                                       const float* __restrict__ X1,
                                       const float* __restrict__ W,
                                       const float* __restrict__ bias,
                                       float* __restrict__ Y, int M) {
  constexpr int CT = NCOLS / 16;   // col tiles per block
  constexpr int RT = 8 / CT;       // row tiles per block
  const int wave = threadIdx.x >> 5;
  const int lane = threadIdx.x & 31;
  const int half = lane >> 4;      // 0/1 -> K sub-offset in A/B frags
  const int lid  = lane & 15;
  const int col0 = (wave % CT) * 16;
  const int row0 = blockIdx.x * (RT * 16) + (wave / CT) * 16;

  const int arow = row0 + lid;                 // A-frag row for this lane
  const int ar   = (arow < M) ? arow : (M - 1);  // clamp (keep EXEC full)

  v8f c = {};
#pragma unroll
  for (int k0 = 0; k0 < K; k0 += 4) {
    const int kb = k0 + half * 2;  // even, never straddles the 64 boundary
    float ax, ay;
    if (SPLIT) {
      ax = (kb < 64) ? X0[ar * 64 + kb]     : X1[ar * 64 + (kb - 64)];
      ay = (kb < 64) ? X0[ar * 64 + kb + 1] : X1[ar * 64 + (kb - 63)];
    } else {
      ax = X0[(size_t)ar * K + kb];
      ay = X0[(size_t)ar * K + kb + 1];
    }
    if (RELUX) { ax = fmaxf(ax, 0.f); ay = fmaxf(ay, 0.f); }
    v2f a; a.x = ax; a.y = ay;
    v2f b;
    b.x = W[(kb)     * NCOLS + col0 + lid];
    b.y = W[(kb + 1) * NCOLS + col0 + lid];
    // (neg_a, A, neg_b, B, c_mod, C, reuse_a, reuse_b)
    c = __builtin_amdgcn_wmma_f32_16x16x4_f32(false, a, false, b, (short)0, c,
                                              false, false);
  }

  const float bv = bias ? bias[col0 + lid] : 0.f;
#pragma unroll
  for (int i = 0; i < 8; ++i) {
    const int row = row0 + i + half * 8;  // VGPR i: M=i (lanes0-15), M=8+i
    if (row < M) Y[(size_t)row * NCOLS + col0 + lid] = c[i] + bv;
  }
}

// ---------------------------------------------------------------------------
// Precompute wv[mat][k][h] = sum_c W[k, h*64+c] * att[h,c]   (3 x 64 x 2)
// mat 0: (W_lin, att_src)  1: (W_lin, att_dst)  2: (W_edge, att_edge)
// ---------------------------------------------------------------------------
__global__ void wv_kernel(const float* __restrict__ W_lin,
                          const float* __restrict__ att_src,
                          const float* __restrict__ att_dst,
                          const float* __restrict__ W_edge,
                          const float* __restrict__ att_edge,
                          float* __restrict__ wv) {
  const int t = threadIdx.x;
  if (t >= 192) return;
  const int mat = t / 64, k = t % 64;
  const float* W = (mat == 2) ? W_edge : W_lin;
  const float* A = (mat == 0) ? att_src : (mat == 1 ? att_dst : att_edge);
  float s0 = 0.f, s1 = 0.f;
  for (int col = 0; col < 64; ++col)  s0 += W[k * 128 + col] * A[col];
  for (int col = 64; col < 128; ++col) s1 += W[k * 128 + col] * A[col];
  wv[mat * 128 + k * 2 + 0] = s0;
  wv[mat * 128 + k * 2 + 1] = s1;
}

// init: conv accumulator seeded with bias_conv; m = -inf; s = 0
__global__ void init_kernel(float* __restrict__ conv, float* __restrict__ m,
                            float* __restrict__ s,
                            const float* __restrict__ bias, int N) {
  const int idx = blockIdx.x * blockDim.x + threadIdx.x;
  if (idx < N * 128) conv[idx] = bias[idx & 127];
  if (idx < N * 2) { m[idx] = -__builtin_inff(); s[idx] = 0.f; }
}

// ---------------------------------------------------------------------------
// a[n,h] = F[n,:64] @ wv[:,h].   Wave handles 2 rows; 16 lanes x float4 = 64.
// ---------------------------------------------------------------------------
__global__ LB256 void logits_kernel(const float* __restrict__ F,
                                    const float* __restrict__ wv,
                                    float* __restrict__ a, int M) {
  const int lane = threadIdx.x & 31;
  const int n = (blockIdx.x * 8 + (threadIdx.x >> 5)) * 2 + (lane >> 4);
  if (n >= M) return;
  const int k4 = (lane & 15) * 4;
  const float4 f = *(const float4*)(F + (size_t)n * 64 + k4);
  float p0 = f.x * wv[(k4 + 0) * 2] + f.y * wv[(k4 + 1) * 2] +
             f.z * wv[(k4 + 2) * 2] + f.w * wv[(k4 + 3) * 2];
  float p1 = f.x * wv[(k4 + 0) * 2 + 1] + f.y * wv[(k4 + 1) * 2 + 1] +
             f.z * wv[(k4 + 2) * 2 + 1] + f.w * wv[(k4 + 3) * 2 + 1];
#pragma unroll
  for (int off = 8; off >= 1; off >>= 1) {
    p0 += __shfl_xor(p0, off, 16);
    p1 += __shfl_xor(p1, off, 16);
  }
  if ((lane & 15) == 0) {
    a[(size_t)n * 2 + 0] = p0;
    a[(size_t)n * 2 + 1] = p1;
  }
}

__device__ inline void atomicMaxF(float* addr, float v) {
  if (v >= 0.f) atomicMax((int*)addr, __float_as_int(v));
  else          atomicMin((unsigned int*)addr, __float_as_uint(v));
}

// alpha = leaky_relu(a_src[src]+a_dst[dst]+a_edge, 0.2); segment max into m
__global__ void alpha_kernel(const int* __restrict__ ei,
                             const float* __restrict__ a_src,
                             const float* __restrict__ a_dst,
                             const float* __restrict__ a_edge,
                             float* __restrict__ alpha, float* __restrict__ m,
                             int E) {
  const int e = blockIdx.x * blockDim.x + threadIdx.x;
  if (e >= E) return;
  const int sn = ei[e], dn = ei[E + e];
#pragma unroll
  for (int h = 0; h < 2; ++h) {
    float al = a_src[sn * 2 + h] + a_dst[dn * 2 + h] + a_edge[(size_t)e * 2 + h];
    al = (al > 0.f) ? al : 0.2f * al;
    alpha[(size_t)e * 2 + h] = al;
    atomicMaxF(&m[dn * 2 + h], al);
  }
}

// ex = exp(alpha - m[dst]) (overwrites alpha); segment sum into s
__global__ void exsum_kernel(const int* __restrict__ ei,
                             float* __restrict__ alpha,
                             const float* __restrict__ m,
                             float* __restrict__ s, int E) {
  const int e = blockIdx.x * blockDim.x + threadIdx.x;
  if (e >= E) return;
  const int dn = ei[E + e];
#pragma unroll
  for (int h = 0; h < 2; ++h) {
    const float ex = expf(alpha[(size_t)e * 2 + h] - m[dn * 2 + h]);
    alpha[(size_t)e * 2 + h] = ex;
    atomicAdd(&s[dn * 2 + h], ex);
  }
}

// att = ex/(s[dst]+1e-16); conv[dst,:] += x_src[src,:] * att[head]
// One wave per edge: lane -> 4 channels (lanes 0-15 head 0, 16-31 head 1).
// x_src (25.6MB) and conv (25.6MB) are L2-resident (192MB L2).
__global__ LB256 void scatter_kernel(const int* __restrict__ ei,
                                     const float* __restrict__ ex,
                                     const float* __restrict__ s,
                                     const float* __restrict__ xsrc,
                                     float* __restrict__ att_out,
                                     float* __restrict__ conv, int E) {
  const int e = blockIdx.x * 8 + (threadIdx.x >> 5);
  if (e >= E) return;
  const int lane = threadIdx.x & 31;
  const int h = lane >> 4;
  const int sn = ei[e], dn = ei[E + e];
  const float att = ex[(size_t)e * 2 + h] / (s[dn * 2 + h] + 1e-16f);
  if ((lane & 15) == 0) att_out[(size_t)e * 2 + h] = att;
  const float4 x4 = *(const float4*)(xsrc + (size_t)sn * 128 + lane * 4);
  float* dp = conv + (size_t)dn * 128 + lane * 4;
  atomicAdd(dp + 0, x4.x * att);
  atomicAdd(dp + 1, x4.y * att);
  atomicAdd(dp + 2, x4.z * att);
  atomicAdd(dp + 3, x4.w * att);
}

// ---------------------------------------------------------------------------
extern "C" void kernel_launch(void* const* d_in, const int* in_sizes, int n_in,
                              void* d_out, int out_size, void* d_ws,
                              size_t ws_size, hipStream_t stream) {
  const float* left  = (const float*)d_in[0];
  const int*   ei    = (const int*)d_in[1];
  const float* efeat = (const float*)d_in[2];
  const float* right = (const float*)d_in[3];
  const float* W_lin = (const float*)d_in[5];
  const float* att_s = (const float*)d_in[6];
  const float* att_d = (const float*)d_in[7];
  const float* W_edg = (const float*)d_in[8];
  const float* att_e = (const float*)d_in[9];
  const float* bconv = (const float*)d_in[10];
  const float* W_lt  = (const float*)d_in[11];
  const float* b_lt  = (const float*)d_in[12];
  const float* W_fm  = (const float*)d_in[13];
  const float* b_fm  = (const float*)d_in[14];
  const float* W_pc  = (const float*)d_in[15];
  const float* b_pc  = (const float*)d_in[16];

  const int N = in_sizes[0] / 64;   // 50000
  const int E = in_sizes[1] / 2;    // 800000

  // workspace layout (floats)
  float* w = (float*)d_ws;
  size_t o = 0;
  float* xsrc   = w + o; o += (size_t)N * 128;
  float* a_src  = w + o; o += (size_t)N * 2;
  float* a_dst  = w + o; o += (size_t)N * 2;
  float* a_edge = w + o; o += (size_t)E * 2;
  float* alpha  = w + o; o += (size_t)E * 2;   // reused for ex
  float* mbuf   = w + o; o += (size_t)N * 2;
  float* sbuf   = w + o; o += (size_t)N * 2;
  float* conv   = w + o; o += (size_t)N * 128;
  float* t1     = w + o; o += (size_t)N * 64;
  float* t2     = w + o; o += (size_t)N * 64;
  float* wv     = w + o; o += 3 * 128;
  (void)ws_size; (void)n_in; (void)out_size;

  // output layout: left_updated [N,64] | edge_index bits [2,E] | att [E,2]
  float* out_lu  = (float*)d_out;
  float* out_ei  = out_lu + (size_t)N * 64;
  float* out_att = out_ei + (size_t)2 * E;

  // 1. fold (W, att) -> 64x2 matvec weights
  wv_kernel<<<1, 192, 0, stream>>>(W_lin, att_s, att_d, W_edg, att_e, wv);
  // 2. init conv accumulator (seed bias_conv), m=-inf, s=0
  init_kernel<<<(N * 128 + 255) / 256, 256, 0, stream>>>(conv, mbuf, sbuf,
                                                         bconv, N);
  // 3. x_src = right @ W_lin   (the one real GEMM; fp32 WMMA)
  gemm_wmma_kernel<64, 128, false, false><<<(N + 15) / 16, 256, 0, stream>>>(
      right, nullptr, W_lin, nullptr, xsrc, N);
  // 4. attention logits as matvecs
  logits_kernel<<<(N + 15) / 16, 256, 0, stream>>>(right, wv + 0,   a_src, N);
  logits_kernel<<<(N + 15) / 16, 256, 0, stream>>>(left,  wv + 128, a_dst, N);
  logits_kernel<<<(E + 15) / 16, 256, 0, stream>>>(efeat, wv + 256, a_edge, E);
  // 5. per-edge logits + segment max
  alpha_kernel<<<(E + 255) / 256, 256, 0, stream>>>(ei, a_src, a_dst, a_edge,
                                                    alpha, mbuf, E);
  // 6. exp + segment sum
  exsum_kernel<<<(E + 255) / 256, 256, 0, stream>>>(ei, alpha, mbuf, sbuf, E);
  // 7. normalize + weighted scatter-add (writes att output too)
  scatter_kernel<<<(E + 7) / 8, 256, 0, stream>>>(ei, alpha, sbuf, xsrc,
                                                  out_att, conv, E);
  // 8. trailing dense chain (fp32 WMMA)
  gemm_wmma_kernel<128, 64, true, false><<<(N + 31) / 32, 256, 0, stream>>>(
      conv, nullptr, W_lt, b_lt, t1, N);
  gemm_wmma_kernel<64, 64, true, false><<<(N + 31) / 32, 256, 0, stream>>>(
      t1, nullptr, W_fm, b_fm, t2, N);
  gemm_wmma_kernel<128, 64, false, true><<<(N + 31) / 32, 256, 0, stream>>>(
      left, t2, W_pc, b_pc, out_lu, N);
  // 9. edge_index passthrough (bit copy into float output region)
  hipMemcpyAsync(out_ei, d_in[1], (size_t)2 * E * sizeof(int),
                 hipMemcpyDeviceToDevice, stream);
}